// nondiff_7421703487559
// MI455X (gfx1250) — compile-verified
//
#include <hip/hip_runtime.h>

// Per-sample QP with the constant KKT solve folded into a WMMA A-operand.
//   z_unc = [x^2, 1]  (M1 solve),  z_act = [-x, 1]  (M2 solve)
//   z     = (-x^2 <= x) ? z_unc : z_act     (ordered <= : NaN -> z_act, matches reference)
//
// V_WMMA_F32_16X16X4_F32:  D = A(16x4) x B(4x16) + 0
//   B column n (K-striped): lane n supplies (K0,K1)=(x^2,x); lane n+16 supplies (K2,K3) of ITS sample
//   A constant: A[0,0]=1, A[1,1]=-1, A[8,2]=1, A[9,3]=-1, rest 0
//   => every lane L ends with D[0] = x_L^2, D[1] = -x_L  (no cross-lane traffic, 32 samples/WMMA)
//
// Bandwidth-bound: 12 B/sample, 96 MB/call -> ~4.1 us at 23.3 TB/s. Hot loop is
// branch-free (scalar trip count), b128 NT loads + 2x b128 NT stores per quad.

typedef __attribute__((ext_vector_type(2))) float v2f;
typedef __attribute__((ext_vector_type(4))) float v4f;
typedef __attribute__((ext_vector_type(8))) float v8f;

#define TPB 256   // 8 waves per block (wave32)
#define EPT 16    // elements per thread (4 quads)

__global__ __launch_bounds__(TPB) void qp_wmma_kernel(const float* __restrict__ x,
                                                      float* __restrict__ out,
                                                      int n) {
  const int lane = (int)(threadIdx.x & 31u);

  // Constant A-matrix operand (16x4 f32, 2 VGPRs per lane):
  //   lanes 0-15 : M=lane,    VGPR0=K0, VGPR1=K1
  //   lanes 16-31: M=lane-16, VGPR0=K2, VGPR1=K3
  v2f A;
  A.x = (lane == 0 || lane == 24) ?  1.0f : 0.0f;   // A[0,0], A[8,2]
  A.y = (lane == 1 || lane == 25) ? -1.0f : 0.0f;   // A[1,1], A[9,3]

  const int t     = (int)(blockIdx.x * TPB + threadIdx.x);
  const int T     = (int)(gridDim.x * TPB);
  const int quads = n >> 2;        // full float4 quads in the input
  const int iters = quads / T;     // SGPR-only -> scalar loop branch, EXEC stays all-1s

  const v4f* __restrict__ x4 = (const v4f*)x;
  v4f* __restrict__ o4       = (v4f*)out;

#pragma unroll 2
  for (int k = 0; k < iters; ++k) {
    const int q = t + k * T;                       // quad index; samples 4q..4q+3
    const v4f xv = __builtin_nontemporal_load(x4 + q);   // 16 B/lane, coalesced, streaming

    v4f z1;
#pragma unroll
    for (int j = 0; j < 4; ++j) {
      const float xj = xv[j];
      v2f B;
      B.x = xj * xj;   // K0 (lanes 0-15) / K2 (lanes 16-31)
      B.y = xj;        // K1 (lanes 0-15) / K3 (lanes 16-31)
      v8f C = {};
      const v8f D = __builtin_amdgcn_wmma_f32_16x16x4_f32(
          /*neg_a=*/false, A, /*neg_b=*/false, B,
          /*c_mod=*/(short)0, C, /*reuse_a=*/false, /*reuse_b=*/false);
      // D[0] = x^2 (rows 0/8), D[1] = -x (rows 1/9)
      z1[j] = (-D[0] <= xj) ? D[0] : D[1];
    }

    v4f o0, o1;
    o0.x = z1[0]; o0.y = 1.0f; o0.z = z1[1]; o0.w = 1.0f;
    o1.x = z1[2]; o1.y = 1.0f; o1.z = z1[3]; o1.w = 1.0f;
    __builtin_nontemporal_store(o0, o4 + 2 * q);       // 32 B/lane, coalesced, streaming
    __builtin_nontemporal_store(o1, o4 + 2 * q + 1);
  }

  // Tail (empty for n = 2^23): scalar math, no WMMA under possibly-partial waves.
  for (int i = iters * T * 4 + t; i < n; i += T) {
    const float xj = x[i];
    const float zu = xj * xj;
    const float z1 = (-zu <= xj) ? zu : -xj;
    v2f o; o.x = z1; o.y = 1.0f;
    __builtin_nontemporal_store(o, (v2f*)out + i);
  }
}

extern "C" void kernel_launch(void* const* d_in, const int* in_sizes, int n_in,
                              void* d_out, int out_size, void* d_ws, size_t ws_size,
                              hipStream_t stream) {
  (void)n_in; (void)out_size; (void)d_ws; (void)ws_size;
  const float* x = (const float*)d_in[0];
  float* out     = (float*)d_out;
  const int n    = in_sizes[0];                      // 8388608

  int blocks = (n + TPB * EPT - 1) / (TPB * EPT);    // 2048 blocks for n = 2^23 (4 passes/thread)
  if (blocks < 1) blocks = 1;

  qp_wmma_kernel<<<blocks, TPB, 0, stream>>>(x, out, n);
}